// HyperGCNConv_21131239096599
// MI455X (gfx1250) — compile-verified
//
#include <hip/hip_runtime.h>

#define NN    100000   // vertices
#define CIN_  128
#define COUT_ 64

typedef __attribute__((ext_vector_type(2))) float v2f;
typedef __attribute__((ext_vector_type(4))) float v4f;
typedef __attribute__((ext_vector_type(8))) float v8f;

// ---------------------------------------------------------------------------
// Zero a buffer of 32-bit words (works for float 0.0f and int 0).
// ---------------------------------------------------------------------------
__global__ void hgcn_zero_kernel(unsigned* __restrict__ p, int n) {
    int i = blockIdx.x * blockDim.x + threadIdx.x;
    if (i < n) p[i] = 0u;
}

// ---------------------------------------------------------------------------
// Degree count: deg[dst[e]] += 1
// ---------------------------------------------------------------------------
__global__ void hgcn_deg_kernel(const int* __restrict__ dst, int* __restrict__ deg, int E) {
    int e = blockIdx.x * blockDim.x + threadIdx.x;
    if (e < E) atomicAdd(&deg[dst[e]], 1);
}

// ---------------------------------------------------------------------------
// dinv[i] = rsqrt(1 + deg[i])
// ---------------------------------------------------------------------------
__global__ void hgcn_dinv_kernel(const int* __restrict__ deg, float* __restrict__ dinv, int n) {
    int i = blockIdx.x * blockDim.x + threadIdx.x;
    if (i < n) dinv[i] = rsqrtf(1.0f + (float)deg[i]);
}

// ---------------------------------------------------------------------------
// H = X @ W + b via v_wmma_f32_16x16x4_f32.
// One wave computes one 16x16 tile of H. 8 waves / block, each wave its own
// M-tile; blockIdx.y selects the N-tile (COUT 64 -> 4 tiles).
//
// A (16x4 f32) lane layout:  lane L holds row M=L&15,  K = (L>>4)*2 + {0,1}
// B (4x16 f32) lane layout:  lane L holds col N=L&15,  K = (L>>4)*2 + {0,1}
// C/D (16x16 f32): 8 VGPRs, lane L holds N=L&15, M = r + (L>=16 ? 8 : 0)
// ---------------------------------------------------------------------------
__global__ __launch_bounds__(256)
void hgcn_gemm_wmma_kernel(const float* __restrict__ X,
                           const float* __restrict__ W,
                           const float* __restrict__ b,
                           float* __restrict__ H) {
    const int lane  = threadIdx.x & 31;
    const int wave  = threadIdx.x >> 5;
    const int mTile = blockIdx.x * 8 + wave;        // 0 .. 6249 (100000/16)
    if (mTile >= (NN / 16)) return;                 // wave-uniform: EXEC stays all-1s
    const int nTile = blockIdx.y;                   // 0 .. 3

    const int l15   = lane & 15;
    const int khalf = (lane >> 4) * 2;              // 0 or 2
    const int row   = mTile * 16 + l15;             // A: M index for this lane
    const int col   = nTile * 16 + l15;             // B/C/D: N index for this lane

    // C initialized with bias (broadcast over the 8 M values this lane holds)
    const float bv = b[col];
    v8f c;
#pragma unroll
    for (int r = 0; r < 8; ++r) c[r] = bv;

    const float* __restrict__ xrow = X + (size_t)row * CIN_;

#pragma unroll 4
    for (int k = 0; k < CIN_; k += 4) {
        v2f a, bb;
        // consecutive in memory -> single b64 load
        a.x  = xrow[k + khalf];
        a.y  = xrow[k + khalf + 1];
        bb.x = W[(size_t)(k + khalf)     * COUT_ + col];
        bb.y = W[(size_t)(k + khalf + 1) * COUT_ + col];
        // 8 args: (neg_a, A, neg_b, B, c_mod, C, reuse_a, reuse_b)
        c = __builtin_amdgcn_wmma_f32_16x16x4_f32(
                false, a, false, bb, (short)0, c, false, false);
    }

    // Scatter the 16x16 f32 tile: lane L writes 8 values at M = mOff + r
    float* __restrict__ hp = H + (size_t)(mTile * 16) * COUT_ + nTile * 16 + l15;
    const int mOff = (lane >> 4) * 8;
#pragma unroll
    for (int r = 0; r < 8; ++r)
        hp[(size_t)(mOff + r) * COUT_] = c[r];
}

// ---------------------------------------------------------------------------
// Edge scatter: 16 threads per edge, one float4 (4 channels) per thread.
// out[dst, c] += H[src, c] * dinv[src] * dinv[dst]
// H (25.6 MB) and out (25.6 MB) both fit in the 192 MB L2 -> atomics stay
// on-chip; HBM only streams the edge lists. float4 gather = global_load_b128.
// ---------------------------------------------------------------------------
__global__ void hgcn_edge_kernel(const int* __restrict__ src,
                                 const int* __restrict__ dst,
                                 const v4f* __restrict__ H4,     // (N, 16) float4
                                 const float* __restrict__ dinv,
                                 float* __restrict__ out, int E) {
    long long t = (long long)blockIdx.x * blockDim.x + threadIdx.x;
    int e = (int)(t >> 4);
    if (e >= E) return;
    int q = (int)(t & 15);                 // channel group: covers channels 4q..4q+3
    int s = src[e];
    int d = dst[e];
    float coef = dinv[s] * dinv[d];
    v4f h = H4[(size_t)s * (COUT_ / 4) + q];
    float* __restrict__ op = out + (size_t)d * COUT_ + q * 4;
    atomicAdd(op + 0, h.x * coef);
    atomicAdd(op + 1, h.y * coef);
    atomicAdd(op + 2, h.z * coef);
    atomicAdd(op + 3, h.w * coef);
}

// ---------------------------------------------------------------------------
// Finalize: out = relu(out + H * dinv^2), float4 vectorized.
// ---------------------------------------------------------------------------
__global__ void hgcn_final_kernel(const v4f* __restrict__ H4,
                                  const float* __restrict__ dinv,
                                  v4f* __restrict__ out4, int total4) {
    int i = blockIdx.x * blockDim.x + threadIdx.x;
    if (i >= total4) return;
    int node = i >> 4;                     // 16 float4s per node (COUT_ == 64)
    float ds = dinv[node];
    float d2 = ds * ds;
    v4f h = H4[i];
    v4f o = out4[i];
    o.x += h.x * d2;  o.y += h.y * d2;  o.z += h.z * d2;  o.w += h.w * d2;
    o.x = o.x > 0.0f ? o.x : 0.0f;
    o.y = o.y > 0.0f ? o.y : 0.0f;
    o.z = o.z > 0.0f ? o.z : 0.0f;
    o.w = o.w > 0.0f ? o.w : 0.0f;
    out4[i] = o;
}

// ---------------------------------------------------------------------------
extern "C" void kernel_launch(void* const* d_in, const int* in_sizes, int n_in,
                              void* d_out, int out_size, void* d_ws, size_t ws_size,
                              hipStream_t stream) {
    const float* X   = (const float*)d_in[0];   // (N, CIN)
    const float* W   = (const float*)d_in[1];   // (CIN, COUT)
    const float* b   = (const float*)d_in[2];   // (COUT,)
    const int*   src = (const int*)d_in[3];     // (E,)
    const int*   dst = (const int*)d_in[4];     // (E,)
    const int    E   = in_sizes[3];

    float* out = (float*)d_out;                 // (N, COUT) — also scatter accumulator

    // Workspace layout: H (N*COUT f32) | deg (N i32) | dinv (N f32)
    char*  ws   = (char*)d_ws;
    float* H    = (float*)ws;
    int*   deg  = (int*)(ws + (size_t)NN * COUT_ * sizeof(float));
    float* dinv = (float*)(ws + (size_t)NN * COUT_ * sizeof(float) + (size_t)NN * sizeof(int));

    const int totalOut = NN * COUT_;

    // 1) zero accumulator (d_out) and degree counts
    hgcn_zero_kernel<<<(totalOut + 255) / 256, 256, 0, stream>>>((unsigned*)out, totalOut);
    hgcn_zero_kernel<<<(NN + 255) / 256, 256, 0, stream>>>((unsigned*)deg, NN);

    // 2) degrees and dinv_sqrt
    hgcn_deg_kernel<<<(E + 255) / 256, 256, 0, stream>>>(dst, deg, E);
    hgcn_dinv_kernel<<<(NN + 255) / 256, 256, 0, stream>>>(deg, dinv, NN);

    // 3) H = X @ W + b  (WMMA f32 16x16x4; 6250 M-tiles x 4 N-tiles)
    dim3 gemmGrid((NN / 16 + 7) / 8, COUT_ / 16);
    hgcn_gemm_wmma_kernel<<<gemmGrid, 256, 0, stream>>>(X, W, b, H);

    // 4) edge scatter-add (16 threads per edge, float4 each)
    long long edgeThreads = (long long)E * 16;
    unsigned edgeBlocks = (unsigned)((edgeThreads + 255) / 256);
    hgcn_edge_kernel<<<edgeBlocks, 256, 0, stream>>>(src, dst, (const v4f*)H, dinv, out, E);

    // 5) self-loop term + ReLU (float4)
    const int total4 = totalOut / 4;
    hgcn_final_kernel<<<(total4 + 255) / 256, 256, 0, stream>>>(
        (const v4f*)H, dinv, (v4f*)out, total4);
}